// feature_extractor_14800457302529
// MI455X (gfx1250) — compile-verified
//
#include <hip/hip_runtime.h>
#include <stdint.h>

// 3x3 zero-padded unfold of two (4,3,384,1248) f32 tensors + constant eye(9).
// 0 FLOPs, ~460 MB HBM traffic -> store-BW bound (~20us @ 23.3 TB/s).
// CDNA5 plan: one TENSOR_LOAD_TO_LDS (TDM, TENSORcnt) DMAs each 10x1248 input
// tile into LDS; the 9 shifted copies stream out as aligned nontemporal b128
// stores fed by shifted LDS reads. Fallback: per-lane async-to-LDS b128 DMA.

typedef __attribute__((ext_vector_type(4))) float v4f;
typedef __attribute__((ext_vector_type(4))) unsigned int u32x4;
typedef __attribute__((ext_vector_type(8))) int i32x8;
typedef __attribute__((ext_vector_type(4))) int i32x4;

#define B_    4
#define C_    3
#define H_    384
#define W_    1248
#define TY    8                       // output rows per block
#define ROWS  (TY + 2)                // staged rows incl. y-halo = 10
#define VPR   (W_ / 4)                // 312 b128 chunks per row
#define NTILE (H_ / TY)               // 48 y-tiles
#define PLANE ((size_t)H_ * (size_t)W_)
#define OUT_T ((size_t)B_ * 27 * (size_t)H_ * (size_t)W_)

__global__ __launch_bounds__(256)
void unfold3x3_tdm_kernel(const float* __restrict__ left,
                          const float* __restrict__ right,
                          float* __restrict__ out)
{
    // contiguous LDS tile: rows r=0..9 hold input rows gy = y0-1+r
    __shared__ alignas(16) float tile[ROWS * W_];   // 49,920 B

    const int tid = threadIdx.x;
    int bid = blockIdx.x;
    const int ytile = bid % NTILE;  bid /= NTILE;
    const int c     = bid % C_;     bid /= C_;
    const int b     = bid % B_;
    const int t     = bid / B_;

    const int y0 = ytile * TY;
    const float* __restrict__ in = t ? right : left;
    const float* plane = in + (size_t)(b * C_ + c) * PLANE;

    // Constant one-hot (identity) filter: eye(9) flattened, diag at idx%10==0.
    if (blockIdx.x == 0 && tid < 81)
        out[2 * OUT_T + (size_t)tid] = (tid % 10 == 0) ? 1.0f : 0.0f;

    // Clamp the DMA'd row range to [0,H); zero-fill the out-of-range halo row.
    const int lds_row0 = (ytile == 0) ? 1 : 0;
    const int gy_start = y0 - 1 + lds_row0;
    const int nrows    = ROWS - lds_row0 - ((ytile == NTILE - 1) ? 1 : 0);

    if (ytile == 0) {
        for (int i = tid; i < VPR; i += 256) {
            v4f z = {0.0f, 0.0f, 0.0f, 0.0f};
            *(v4f*)&tile[4 * i] = z;                       // LDS row 0 (y = -1)
        }
    }
    if (ytile == NTILE - 1) {
        for (int i = tid; i < VPR; i += 256) {
            v4f z = {0.0f, 0.0f, 0.0f, 0.0f};
            *(v4f*)&tile[(ROWS - 1) * W_ + 4 * i] = z;     // LDS row 9 (y = H)
        }
    }

#if __has_builtin(__builtin_amdgcn_tensor_load_to_lds)
    // ---- Tensor Data Mover: one descriptor DMAs the whole tile ----
    if ((tid >> 5) == 0) {   // wave 0 only (EXEC ignored; one issue per wave)
        const uint64_t ga = (uint64_t)(uintptr_t)(plane + (size_t)gy_start * W_);
        const uint32_t la = (uint32_t)(uintptr_t)&tile[0]
                          + (uint32_t)(lds_row0 * W_ * 4);
        u32x4 g0;
        g0.x = 1u;                                          // count=1 (valid D#)
        g0.y = la;                                          // lds_addr (bytes)
        g0.z = (uint32_t)ga;                                // global_addr[31:0]
        g0.w = (uint32_t)((ga >> 32) & 0x01FFFFFFu)         // global_addr[56:32]
             | (2u << 30);                                  // type=2 ("image")
        i32x8 g1;
        g1[0] = 0x00020000;                                 // data_size=2 (4 B)
        g1[1] = (W_ & 0xFFFF) << 16;                        // tensor_dim0 = W
        g1[2] = (nrows & 0xFFFF) << 16;                     // tensor_dim1 = nrows
        g1[3] = (W_ & 0xFFFF) << 16;                        // tile_dim0 = W
        g1[4] = (nrows & 0xFFFF);                           // tile_dim1 = nrows
        g1[5] = W_;                                         // tensor_dim0_stride
        g1[6] = 0;
        g1[7] = 0;
        i32x4 z4 = {0, 0, 0, 0};                            // groups 2/3 unused (2-D)
        i32x8 z8 = {0, 0, 0, 0, 0, 0, 0, 0};
        __builtin_amdgcn_tensor_load_to_lds(g0, g1, z4, z4, z8, 0);
        __builtin_amdgcn_s_wait_tensorcnt(0);
    }
#else
    // ---- Fallback: per-lane async-DMA b128 chunks into LDS (ASYNCcnt) ----
    {
        const uint32_t lds_base = (uint32_t)(uintptr_t)&tile[0];
        const uint64_t sbase = (uint64_t)(uintptr_t)(plane + (size_t)gy_start * W_);
        for (int i = tid; i < nrows * VPR; i += 256) {
            const int r = i / VPR;
            const int v = i - r * VPR;
            const uint32_t lofs = lds_base + (uint32_t)(((lds_row0 + r) * W_ + 4 * v) * 4);
            const uint32_t gofs = (uint32_t)((r * W_ + 4 * v) * 4);
            asm volatile("global_load_async_to_lds_b128 %0, %1, %2"
                         :: "v"(lofs), "v"(gofs), "s"(sbase)
                         : "memory");
        }
        asm volatile("s_wait_asynccnt 0" ::: "memory");
    }
#endif
    __syncthreads();

    // ---- 9 shifted copies: aligned nontemporal b128 stores from LDS ----
    float* outT = out + (size_t)t * OUT_T;
    const int total4 = 9 * TY * VPR;              // 22,464 b128 stores per block
    for (int idx = tid; idx < total4; idx += 256) {
        int k   = idx / (TY * VPR);
        int rem = idx - k * (TY * VPR);
        int ly  = rem / VPR;
        int j   = rem - ly * VPR;
        const int dy = k / 3;
        const int dx = k - 3 * dy;
        const int r = ly + dy;
        const int xbase = 4 * j + dx - 1;         // input col of element 0

        v4f val;
        if (j == 0 || j == VPR - 1) {
            // edge chunks: clamp x = -1 / x = W to the zero padding
#pragma unroll
            for (int e = 0; e < 4; ++e) {
                const int xc = xbase + e;
                val[e] = ((unsigned)xc < (unsigned)W_) ? tile[r * W_ + xc] : 0.0f;
            }
        } else {
            const float* src = &tile[r * W_ + xbase];
            val.x = src[0];
            val.y = src[1];
            val.z = src[2];
            val.w = src[3];
        }

        const size_t off =
            ((size_t)(b * 27 + (k * 3 + c)) * H_ + (size_t)(y0 + ly)) * W_ + (size_t)(4 * j);
        __builtin_nontemporal_store(val, (v4f*)(outT + off));
    }
}

extern "C" void kernel_launch(void* const* d_in, const int* in_sizes, int n_in,
                              void* d_out, int out_size, void* d_ws, size_t ws_size,
                              hipStream_t stream) {
    (void)in_sizes; (void)n_in; (void)out_size; (void)d_ws; (void)ws_size;
    const float* left  = (const float*)d_in[0];
    const float* right = (const float*)d_in[1];
    float* out = (float*)d_out;

    dim3 grid(2 * B_ * C_ * NTILE);   // 1152 blocks: (tensor, b, c, y-tile)
    dim3 block(256);                  // 8 wave32 waves
    unfold3x3_tdm_kernel<<<grid, block, 0, stream>>>(left, right, out);
}